// QuantumCell_33887291965880
// MI455X (gfx1250) — compile-verified
//
#include <hip/hip_runtime.h>

// ---------------------------------------------------------------------------
// Batched complex 64x64 expm (Pade-13, 1 squaring) + state application.
// out = expm(-i*dt*H) @ state,  B=4096, d=64, complex64.
// One workgroup (512 threads = 16 wave32) per batch matrix; all GEMMs via
// v_wmma_f32_16x16x4_f32. (V-U)^{-1}(V+U) solved with Newton-Schulz since
// V-U = b0(I+E), ||E||~0.3.
// LDS: 5 complex 64x64 buffers (10 fp32 planes) = 163,840 B -> 2 WGs / WGP.
// ---------------------------------------------------------------------------

typedef __attribute__((ext_vector_type(2))) float v2f;
typedef __attribute__((ext_vector_type(8))) float v8f;

#define NT    512
#define PLANE 4096   // 64*64 floats per plane

// Pade-13 coefficients (scipy/tf expm degree-13)
#define PB0  64764752532480000.0f
#define PB1  32382376266240000.0f
#define PB2  7771770303897600.0f
#define PB3  1187353796428800.0f
#define PB4  129060195264000.0f
#define PB5  10559470521600.0f
#define PB6  670442572800.0f
#define PB7  33522128640.0f
#define PB8  1323241920.0f
#define PB9  40840800.0f
#define PB10 960960.0f
#define PB11 16380.0f
#define PB12 182.0f
#define PB13 1.0f

// XOR swizzle: conflict-free LDS banks for both row and column tile access.
__device__ __forceinline__ int sw(int i, int j) { return (i << 6) | ((j ^ i) & 63); }

__device__ __forceinline__ v8f wmma4(v2f a, v2f b, v8f c) {
  // D = A(16x4,f32) x B(4x16,f32) + C(16x16,f32)
  return __builtin_amdgcn_wmma_f32_16x16x4_f32(false, a, false, b, (short)0, c,
                                               false, false);
}

struct Geom { int r, c, half, l; };  // tile row/col base, lane-half, lane-in-half

// C = A @ B  (complex, 64x64, all operands in swizzled LDS planes)
__device__ __forceinline__ void cgemm(const float* __restrict__ Ar, const float* __restrict__ Ai,
                                      const float* __restrict__ Br, const float* __restrict__ Bi,
                                      float* __restrict__ Cr, float* __restrict__ Ci, Geom g) {
  v8f rr = {}, ii = {}, ri = {}, ir = {};
  const int row = g.r + g.l;
  const int col = g.c + g.l;
#pragma unroll 4
  for (int k = 0; k < 64; k += 4) {
    const int ka = k + g.half * 2;     // A: lanes 0-15 -> K=k..k+1, lanes 16-31 -> K=k+2..k+3
    v2f ar, ai, br, bi;
    ar[0] = Ar[sw(row, ka)];      ar[1] = Ar[sw(row, ka + 1)];
    ai[0] = Ai[sw(row, ka)];      ai[1] = Ai[sw(row, ka + 1)];
    br[0] = Br[sw(ka, col)];      br[1] = Br[sw(ka + 1, col)];
    bi[0] = Bi[sw(ka, col)];      bi[1] = Bi[sw(ka + 1, col)];
    rr = wmma4(ar, br, rr);  ii = wmma4(ai, bi, ii);
    ri = wmma4(ar, bi, ri);  ir = wmma4(ai, br, ir);
  }
#pragma unroll
  for (int q = 0; q < 8; ++q) {
    const int m = g.r + q + g.half * 8;
    Cr[sw(m, col)] = rr[q] - ii[q];
    Ci[sw(m, col)] = ri[q] + ir[q];
  }
}

// C = A6 @ (kb3*A6 + kb2*A4 + kb1*A2)  +  ka3*A6 + ka2*A4 + ka1*A2 + kI*I
// (B-operand scaling and additive terms fused; saves two 32KB LDS buffers)
__device__ __forceinline__ void cgemm_poly(
    const float* __restrict__ A6r, const float* __restrict__ A6i,
    const float* __restrict__ A4r, const float* __restrict__ A4i,
    const float* __restrict__ A2r, const float* __restrict__ A2i,
    float kb3, float kb2, float kb1,
    float ka3, float ka2, float ka1, float kI,
    float* __restrict__ Cr, float* __restrict__ Ci, Geom g) {
  v8f rr = {}, ii = {}, ri = {}, ir = {};
  const int row = g.r + g.l;
  const int col = g.c + g.l;
#pragma unroll 2
  for (int k = 0; k < 64; k += 4) {
    const int ka = k + g.half * 2;
    v2f ar, ai, br, bi;
#pragma unroll
    for (int u = 0; u < 2; ++u) {
      const int oa = sw(row, ka + u);
      ar[u] = A6r[oa];  ai[u] = A6i[oa];
      const int ob = sw(ka + u, col);
      br[u] = kb3 * A6r[ob] + kb2 * A4r[ob] + kb1 * A2r[ob];
      bi[u] = kb3 * A6i[ob] + kb2 * A4i[ob] + kb1 * A2i[ob];
    }
    rr = wmma4(ar, br, rr);  ii = wmma4(ai, bi, ii);
    ri = wmma4(ar, bi, ri);  ir = wmma4(ai, br, ir);
  }
#pragma unroll
  for (int q = 0; q < 8; ++q) {
    const int m = g.r + q + g.half * 8;
    const int o = sw(m, col);
    const float dg = (m == col) ? kI : 0.0f;
    Cr[o] = rr[q] - ii[q] + ka3 * A6r[o] + ka2 * A4r[o] + ka1 * A2r[o] + dg;
    Ci[o] = ri[q] + ir[q] + ka3 * A6i[o] + ka2 * A4i[o] + ka1 * A2i[o];
  }
}

__global__ void __launch_bounds__(NT)
quantum_expm_kernel(const float* __restrict__ Hre, const float* __restrict__ Him,
                    const float* __restrict__ Sre, const float* __restrict__ Sim,
                    float* __restrict__ Out) {
  __shared__ float lds[10 * PLANE];   // 163,840 B: 5 complex 64x64 buffers
  float* b0r = lds + 0 * PLANE;  float* b0i = lds + 1 * PLANE;
  float* b1r = lds + 2 * PLANE;  float* b1i = lds + 3 * PLANE;
  float* b2r = lds + 4 * PLANE;  float* b2i = lds + 5 * PLANE;
  float* b3r = lds + 6 * PLANE;  float* b3i = lds + 7 * PLANE;
  float* b4r = lds + 8 * PLANE;  float* b4i = lds + 9 * PLANE;

  const int tid  = threadIdx.x;
  const int wave = tid >> 5;
  const int lane = tid & 31;
  Geom g{ (wave >> 2) << 4, (wave & 3) << 4, lane >> 4, lane & 15 };
  const size_t base = (size_t)blockIdx.x * PLANE;
  const float  sc   = 0.5f * 0.05f;   // 0.5 (scaling step) * delta_T

  // b0 = A = 0.5*dt*(-i)*H : re = sc*Him, im = -sc*Hre (swizzled store)
  for (int t = tid; t < PLANE; t += NT) {
    const int o = (t & 0xFC0) | (((t >> 6) ^ t) & 63);
    b0r[o] =  sc * Him[base + t];
    b0i[o] = -sc * Hre[base + t];
  }
  __syncthreads();
  cgemm(b0r, b0i, b0r, b0i, b1r, b1i, g);                 // b1 = A2
  __syncthreads();
  cgemm(b1r, b1i, b1r, b1i, b2r, b2i, g);                 // b2 = A4
  __syncthreads();
  cgemm(b1r, b1i, b2r, b2i, b3r, b3i, g);                 // b3 = A6
  __syncthreads();
  // b4 = V = A6@(b12 A6 + b10 A4 + b8 A2) + b6 A6 + b4 A4 + b2 A2 + b0 I
  cgemm_poly(b3r, b3i, b2r, b2i, b1r, b1i,
             PB12, PB10, PB8, PB6, PB4, PB2, PB0, b4r, b4i, g);
  __syncthreads();
  // b0 = Uin = A6@(b13 A6 + b11 A4 + b9 A2) + b7 A6 + b5 A4 + b3 A2 + b1 I
  // (overwrites A; A is reloaded from H below — cheap vs an extra LDS buffer)
  cgemm_poly(b3r, b3i, b2r, b2i, b1r, b1i,
             PB13, PB11, PB9, PB7, PB5, PB3, PB1, b0r, b0i, g);
  __syncthreads();
  // b1 = A (reload)
  for (int t = tid; t < PLANE; t += NT) {
    const int o = (t & 0xFC0) | (((t >> 6) ^ t) & 63);
    b1r[o] =  sc * Him[base + t];
    b1i[o] = -sc * Hre[base + t];
  }
  __syncthreads();
  cgemm(b1r, b1i, b0r, b0i, b2r, b2i, g);                 // b2 = U = A @ Uin
  __syncthreads();
  // b0 = N = V - U ; b3 = P = V + U
  for (int o = tid; o < PLANE; o += NT) {
    const float vr = b4r[o], vi = b4i[o], ur = b2r[o], ui = b2i[o];
    b0r[o] = vr - ur;  b0i[o] = vi - ui;
    b3r[o] = vr + ur;  b3i[o] = vi + ui;
  }
  __syncthreads();
  // Newton-Schulz inverse of N: X0 = (1/b0)(2I - N/b0) (first step analytic)
  const float cb = 1.0f / PB0;
  for (int o = tid; o < PLANE; o += NT) {
    const float dg = ((o & 63) == 0) ? 2.0f : 0.0f;   // swizzled diagonal
    b2r[o] = cb * (dg - cb * b0r[o]);
    b2i[o] = -cb * cb * b0i[o];
  }
  __syncthreads();
  float *Xr = b2r, *Xi = b2i, *Zr = b1r, *Zi = b1i;
  for (int it = 0; it < 5; ++it) {                        // quadratic: err -> err^2
    cgemm(b0r, b0i, Xr, Xi, b4r, b4i, g);                 // T = N @ X
    __syncthreads();
    for (int o = tid; o < PLANE; o += NT) {               // Y = 2I - T (in place)
      const float dg = ((o & 63) == 0) ? 2.0f : 0.0f;
      b4r[o] = dg - b4r[o];
      b4i[o] = -b4i[o];
    }
    __syncthreads();
    cgemm(Xr, Xi, b4r, b4i, Zr, Zi, g);                   // Xn = X @ Y
    __syncthreads();
    float* t1 = Xr; Xr = Zr; Zr = t1;
    t1 = Xi; Xi = Zi; Zi = t1;
  }
  cgemm(Xr, Xi, b3r, b3i, b4r, b4i, g);                   // b4 = R = N^-1 @ P
  __syncthreads();
  cgemm(b4r, b4i, b4r, b4i, b0r, b0i, g);                 // b0 = R@R (1 squaring)
  __syncthreads();
  // b3 = state
  for (int t = tid; t < PLANE; t += NT) {
    const int o = (t & 0xFC0) | (((t >> 6) ^ t) & 63);
    b3r[o] = Sre[base + t];
    b3i[o] = Sim[base + t];
  }
  __syncthreads();
  // out = R2 @ state, accumulators -> global directly
  {
    v8f rr = {}, ii = {}, ri = {}, ir = {};
    const int row = g.r + g.l;
    const int col = g.c + g.l;
#pragma unroll 4
    for (int k = 0; k < 64; k += 4) {
      const int ka = k + g.half * 2;
      v2f ar, ai, br, bi;
      ar[0] = b0r[sw(row, ka)];   ar[1] = b0r[sw(row, ka + 1)];
      ai[0] = b0i[sw(row, ka)];   ai[1] = b0i[sw(row, ka + 1)];
      br[0] = b3r[sw(ka, col)];   br[1] = b3r[sw(ka + 1, col)];
      bi[0] = b3i[sw(ka, col)];   bi[1] = b3i[sw(ka + 1, col)];
      rr = wmma4(ar, br, rr);  ii = wmma4(ai, bi, ii);
      ri = wmma4(ar, bi, ri);  ir = wmma4(ai, br, ir);
    }
    const size_t imOff = (size_t)gridDim.x * PLANE;
#pragma unroll
    for (int q = 0; q < 8; ++q) {
      const int m = g.r + q + g.half * 8;
      Out[base + (size_t)m * 64 + col]         = rr[q] - ii[q];
      Out[imOff + base + (size_t)m * 64 + col] = ri[q] + ir[q];
    }
  }
}

extern "C" void kernel_launch(void* const* d_in, const int* in_sizes, int n_in,
                              void* d_out, int out_size, void* d_ws, size_t ws_size,
                              hipStream_t stream) {
  (void)n_in; (void)out_size; (void)d_ws; (void)ws_size;
  const float* Hre = (const float*)d_in[0];
  const float* Him = (const float*)d_in[1];
  const float* Sre = (const float*)d_in[2];
  const float* Sim = (const float*)d_in[3];
  float* out = (float*)d_out;
  const int B = in_sizes[0] >> 12;   // elements per 64x64 matrix = 4096
  hipLaunchKernelGGL(quantum_expm_kernel, dim3(B), dim3(NT), 0, stream,
                     Hre, Him, Sre, Sim, out);
}